// DualAttn_71073118814825
// MI455X (gfx1250) — compile-verified
//
#include <hip/hip_runtime.h>

typedef __attribute__((ext_vector_type(2))) float v2f;
typedef __attribute__((ext_vector_type(8))) float v8f;

#define BB 8
#define CIN 256
#define LIN 2046
#define LP 2048
#define IC 64
#define QC 8
#define OCH 128
#define BN_EPS 1e-5f

// ---------------- init: zero pad columns, pack Wq|Wk -> Wqk, bq|bk -> bqk ----
__global__ void init_pack_kernel(const float* __restrict__ Wq, const float* __restrict__ Wk,
                                 const float* __restrict__ bq, const float* __restrict__ bk,
                                 float* __restrict__ ypad1, float* __restrict__ ypad2,
                                 float* __restrict__ Wqk, float* __restrict__ bqk) {
  int idx = blockIdx.x * blockDim.x + threadIdx.x;
  if (idx < 1024) {                     // 8 batches * 64 ch * 2 pad cols
    int b = idx >> 7, rem = idx & 127;
    int ch = rem >> 1;
    int l = (rem & 1) ? (LP - 1) : 0;
    size_t off = ((size_t)(b * IC + ch)) * LP + l;
    ypad1[off] = 0.f; ypad2[off] = 0.f;
  } else if (idx < 1536) {
    int j = idx - 1024; Wqk[j] = Wq[j];          // 8x64
  } else if (idx < 2048) {
    int j = idx - 1536; Wqk[512 + j] = Wk[j];    // 8x64
  } else if (idx < 2064) {
    int j = idx - 2048; bqk[j] = (j < 8) ? bq[j] : bk[j - 8];
  }
}

// ---------------- generic pointwise conv (matmul over positions) via WMMA ----
// Y[b, o, n+padoff] = sum_c W[o,c] * X[b,c,n] (+ bias[o])
// grid: x = ceil(Lin/64) (4 waves * 16 cols), y = O/16, z = B ; block = 128
// Out-of-range columns load from a clamped address (garbage only pollutes the
// same D column, whose store is guarded), so the K-loop is branch-free.
__global__ void conv1x1_wmma_kernel(const float* __restrict__ X, const float* __restrict__ W,
                                    const float* __restrict__ bias, float* __restrict__ Y,
                                    int Cin, int Lin, int Lx, int Ly, int padoff) {
  int wave = threadIdx.x >> 5;
  int lane = threadIdx.x & 31;
  int row  = lane & 15;          // M index for A, N index for B/D
  int hi   = lane >> 4;          // half-wave selector (K offset 0 or 2)
  int O = gridDim.y * 16;
  int b = blockIdx.z;
  int obase = blockIdx.y * 16;
  int nbase = blockIdx.x * 64 + wave * 16;
  int n = nbase + row;
  bool valid = (n < Lin);
  int nc = valid ? n : (Lin - 1);                // clamped, always in-bounds
  const float* xp = X + (size_t)b * Cin * Lx + (size_t)(2 * hi) * Lx + nc;
  const float* wp = W + (size_t)(obase + row) * Cin + 2 * hi;
  const size_t xstep = (size_t)4 * Lx;
  v8f acc = {0.f, 0.f, 0.f, 0.f, 0.f, 0.f, 0.f, 0.f};
  #pragma unroll 4
  for (int k = 0; k < Cin; k += 4) {
    v2f a, bm;
    a.x  = wp[0];
    a.y  = wp[1];
    bm.x = xp[0];
    bm.y = xp[Lx];
    acc = __builtin_amdgcn_wmma_f32_16x16x4_f32(false, a, false, bm, (short)0, acc, false, false);
    wp += 4;
    xp += xstep;
  }
  if (valid) {
    size_t ybase = (size_t)b * O * Ly + (size_t)n + padoff;
    #pragma unroll
    for (int r = 0; r < 8; ++r) {
      int o = obase + r + 8 * hi;
      float val = acc[r];
      if (bias) val += bias[o];
      Y[ybase + (size_t)o * Ly] = val;
    }
  }
}

// ---------------- BatchNorm stats (training mode, biased var) ----------------
// one block per channel; writes coef[ch*2] = g*rsqrt(var+eps), coef[ch*2+1] = b - mu*a
__global__ void bn_stats_kernel(const float* __restrict__ Y, const float* __restrict__ g,
                                const float* __restrict__ bb, float* __restrict__ coef) {
  __shared__ float ss[256], sq[256];
  int ch = blockIdx.x;
  int tid = threadIdx.x;
  float s = 0.f, q = 0.f;
  // iterate float4-wide: BB*LP/4 = 4096 vectors per channel
  for (int i = tid; i < (BB * LP) / 4; i += 256) {
    int batch = i >> 9;                          // 512 float4 per (batch,channel) row
    int l4 = i & 511;
    const float4 v = *(const float4*)(Y + ((size_t)(batch * IC + ch)) * LP + (size_t)l4 * 4);
    s += v.x + v.y + v.z + v.w;
    q += v.x * v.x + v.y * v.y + v.z * v.z + v.w * v.w;
  }
  ss[tid] = s; sq[tid] = q; __syncthreads();
  for (int o = 128; o > 0; o >>= 1) {
    if (tid < o) { ss[tid] += ss[tid + o]; sq[tid] += sq[tid + o]; }
    __syncthreads();
  }
  if (tid == 0) {
    float N = (float)(BB * LP);
    float mu = ss[0] / N;
    float var = sq[0] / N - mu * mu;
    float inv = rsqrtf(var + BN_EPS);
    float a = g[ch] * inv;
    coef[ch * 2] = a;
    coef[ch * 2 + 1] = bb[ch] - mu * a;
  }
}

// float4 elementwise BN+ReLU; grid covers BB*IC*LP/4 vectors
__global__ void bn_apply_kernel(const float* __restrict__ Yin, const float* __restrict__ coef,
                                float* __restrict__ Yout) {
  size_t i4 = (size_t)blockIdx.x * 256 + threadIdx.x;  // float4 index
  int ch = (int)((i4 >> 9) & (IC - 1));                // 512 float4 per channel row
  float a = coef[ch * 2], c = coef[ch * 2 + 1];
  float4 v = *(const float4*)(Yin + i4 * 4);
  v.x = fmaxf(v.x * a + c, 0.f);
  v.y = fmaxf(v.y * a + c, 0.f);
  v.z = fmaxf(v.z * a + c, 0.f);
  v.w = fmaxf(v.w * a + c, 0.f);
  *(float4*)(Yout + i4 * 4) = v;
}

// ---------------- fused flash-style PAM (softmax over m, WMMA v*p^T) ---------
// grid: (LP/16, B), block 128 (4 waves, each owns a 16ch x 16l output tile)
__global__ void pam_flash_kernel(const float* __restrict__ qk, const float* __restrict__ vbuf,
                                 const float* __restrict__ feat1, const float* __restrict__ gamma,
                                 float* __restrict__ sa) {
  __shared__ float q_s[QC * 16];
  __shared__ float p_s[16 * 16];
  __shared__ float mrun[16], srun[16], scl[16];
  int tid = threadIdx.x;
  int wave = tid >> 5, lane = tid & 31, row = lane & 15, hi = lane >> 4;
  int b = blockIdx.y, lbase = blockIdx.x * 16;
  const float* qb = qk + (size_t)b * 16 * LP;     // rows 0..7 = q
  const float* kb = qb + 8 * LP;                  // rows 8..15 = k
  const float* vb = vbuf + (size_t)b * IC * LP;
  int cbase = wave * 16;
  const float* vrow = vb + (size_t)(cbase + row) * LP;   // per-lane V row
  {
    int ch = tid >> 4, nn = tid & 15;             // 128 threads = 8ch x 16l
    q_s[tid] = qb[(size_t)ch * LP + lbase + nn];
  }
  if (tid < 16) { mrun[tid] = -1e30f; srun[tid] = 0.f; }
  __syncthreads();
  v8f acc = {0.f, 0.f, 0.f, 0.f, 0.f, 0.f, 0.f, 0.f};
  for (int mt = 0; mt < LP / 16; ++mt) {
    int mbase = mt * 16;
    // prefetch next tile's K and V data (speculative, gfx1250 global_prefetch)
    int mnext = mbase + 16;
    if (mnext < LP) {
      __builtin_prefetch(&kb[(size_t)(tid >> 4) * LP + mnext + (tid & 15)], 0, 0);
      if (row == 0) __builtin_prefetch(vrow + mnext, 0, 0);
    }
    // energy tile e[l][j] = sum_ch q[ch,l]*k[ch,mbase+j]
    for (int e = tid; e < 256; e += 128) {
      int l = e >> 4, j = e & 15;
      float s = 0.f;
      #pragma unroll
      for (int ch = 0; ch < QC; ++ch) s += q_s[ch * 16 + l] * kb[(size_t)ch * LP + mbase + j];
      p_s[e] = s;
    }
    __syncthreads();
    if (tid < 16) {                               // per-row running max + rescale factor
      float tmax = p_s[tid * 16];
      for (int j = 1; j < 16; ++j) tmax = fmaxf(tmax, p_s[tid * 16 + j]);
      float mnew = fmaxf(mrun[tid], tmax);
      scl[tid] = expf(mrun[tid] - mnew);
      mrun[tid] = mnew;
    }
    __syncthreads();
    for (int e = tid; e < 256; e += 128) {        // p = exp(e - m_new)
      int l = e >> 4;
      p_s[e] = expf(p_s[e] - mrun[l]);
    }
    __syncthreads();
    if (tid < 16) {                               // running denominator
      float rs = 0.f;
      for (int j = 0; j < 16; ++j) rs += p_s[tid * 16 + j];
      srun[tid] = srun[tid] * scl[tid] + rs;
    }
    acc = acc * scl[row];                         // rescale accumulator columns (N=row)
    #pragma unroll
    for (int ks = 0; ks < 4; ++ks) {              // D += V(16c x 4m) * P^T(4m x 16l)
      v2f a, bm;
      int mk = mbase + ks * 4 + 2 * hi;
      a.x = vrow[mk];
      a.y = vrow[mk + 1];
      bm.x = p_s[row * 16 + ks * 4 + 2 * hi];
      bm.y = p_s[row * 16 + ks * 4 + 2 * hi + 1];
      acc = __builtin_amdgcn_wmma_f32_16x16x4_f32(false, a, false, bm, (short)0, acc, false, false);
    }
    __syncthreads();
  }
  float gm = gamma[0];
  int l = lbase + row;
  float inv = 1.f / srun[row];
  #pragma unroll
  for (int r = 0; r < 8; ++r) {
    int c = cbase + r + 8 * hi;
    size_t idx = ((size_t)(b * IC + c)) * LP + l;
    sa[idx] = gm * (acc[r] * inv) + feat1[idx];
  }
}

// ---------------- CAM: channel Gram matrix, softmax, apply ------------------
__global__ void cam_energy_kernel(const float* __restrict__ f2, float* __restrict__ E) {
  __shared__ float Fs[IC * 128];
  int b = blockIdx.y;
  int ctile = blockIdx.x * 16;
  int tid = threadIdx.x;
  float acc[4] = {0.f, 0.f, 0.f, 0.f};
  for (int lc = 0; lc < LP; lc += 128) {
    // stage 64ch x 128pos tile as float4 (2048 vectors / 256 threads = 8 each)
    for (int i = tid; i < (IC * 128) / 4; i += 256) {
      int ch = i >> 5, pos4 = i & 31;
      float4 v = *(const float4*)(f2 + ((size_t)(b * IC + ch)) * LP + lc + (size_t)pos4 * 4);
      *(float4*)(Fs + ch * 128 + pos4 * 4) = v;
    }
    __syncthreads();
    #pragma unroll
    for (int r = 0; r < 4; ++r) {
      int e = tid + 256 * r;
      int c = ctile + (e >> 6), d = e & 63;
      float s = acc[r];
      for (int p = 0; p < 128; ++p) s += Fs[c * 128 + p] * Fs[d * 128 + p];
      acc[r] = s;
    }
    __syncthreads();
  }
  #pragma unroll
  for (int r = 0; r < 4; ++r) {
    int e = tid + 256 * r;
    int c = ctile + (e >> 6), d = e & 63;
    E[(size_t)b * IC * IC + c * IC + d] = acc[r];
  }
}

__global__ void cam_softmax_kernel(float* __restrict__ E) {
  int b = blockIdx.x, c = threadIdx.x;
  float* row = E + (size_t)b * IC * IC + (size_t)c * IC;
  float rmax = row[0];
  for (int d = 1; d < IC; ++d) rmax = fmaxf(rmax, row[d]);
  float m2 = rmax - row[0];                 // max of energy_new = rmax - e
  for (int d = 1; d < IC; ++d) m2 = fmaxf(m2, rmax - row[d]);
  float sum = 0.f;
  for (int d = 0; d < IC; ++d) {
    float p = expf((rmax - row[d]) - m2);
    row[d] = p; sum += p;
  }
  float inv = 1.f / sum;
  for (int d = 0; d < IC; ++d) row[d] *= inv;
}

__global__ void cam_out_kernel(const float* __restrict__ E, const float* __restrict__ f2,
                               const float* __restrict__ gamma, float* __restrict__ sc) {
  size_t i = (size_t)blockIdx.x * 256 + threadIdx.x;   // BB*IC*LP
  int l = (int)(i & (LP - 1));
  int c = (int)((i >> 11) & (IC - 1));
  int b = (int)(i >> 17);
  const float* att = E + (size_t)b * IC * IC + (size_t)c * IC;
  const float* fb = f2 + (size_t)b * IC * LP + l;
  float s = 0.f;
  for (int d = 0; d < IC; ++d) s += att[d] * fb[(size_t)d * LP];
  sc[i] = gamma[0] * s + f2[i];
}

// ---------------- dense conv k=3 pad=1 --------------------------------------
__global__ void conv3_kernel(const float* __restrict__ F, const float* __restrict__ W,
                             float* __restrict__ Z) {
  __shared__ float Ws[IC * 3];
  int tid = threadIdx.x;
  int o = blockIdx.y, b = blockIdx.z;
  int l = blockIdx.x * 256 + tid;
  if (tid < IC * 3) Ws[tid] = W[o * IC * 3 + tid];
  __syncthreads();
  const float* fb = F + (size_t)b * IC * LP + l;
  float acc = 0.f;
  bool lok = (l >= 1), rok = (l <= LP - 2);
  for (int i = 0; i < IC; ++i) {
    const float* base = fb + (size_t)i * LP;
    float x0 = lok ? base[-1] : 0.f;
    float x1 = base[0];
    float x2 = rok ? base[1] : 0.f;
    acc += Ws[i * 3] * x0 + Ws[i * 3 + 1] * x1 + Ws[i * 3 + 2] * x2;
  }
  Z[((size_t)(b * IC + o)) * LP + l] = acc;
}

// ---------------- sum over L of (sa_conv + sc_conv) and final head ----------
__global__ void sum_l_kernel(const float* __restrict__ A, const float* __restrict__ Bm,
                             float* __restrict__ S) {
  __shared__ float ss[256];
  int bi = blockIdx.x;                   // b*IC + i
  int tid = threadIdx.x;
  float s = 0.f;
  size_t base = (size_t)bi * LP;
  for (int l4 = tid; l4 < LP / 4; l4 += 256) {
    float4 va = *(const float4*)(A + base + (size_t)l4 * 4);
    float4 vb = *(const float4*)(Bm + base + (size_t)l4 * 4);
    s += (va.x + vb.x) + (va.y + vb.y) + (va.z + vb.z) + (va.w + vb.w);
  }
  ss[tid] = s; __syncthreads();
  for (int o = 128; o > 0; o >>= 1) { if (tid < o) ss[tid] += ss[tid + o]; __syncthreads(); }
  if (tid == 0) S[bi] = ss[0];
}

__global__ void final_kernel(const float* __restrict__ W8, const float* __restrict__ b8,
                             const float* __restrict__ S, float* __restrict__ out) {
  int b = blockIdx.x, o = threadIdx.x;
  float s = 0.f;
  for (int i = 0; i < IC; ++i) s += W8[o * IC + i] * S[b * IC + i];
  out[b * OCH + o] = s / (float)LP + b8[o];
}

extern "C" void kernel_launch(void* const* d_in, const int* in_sizes, int n_in,
                              void* d_out, int out_size, void* d_ws, size_t ws_size,
                              hipStream_t stream) {
  (void)in_sizes; (void)n_in; (void)out_size; (void)ws_size;
  const float* x    = (const float*)d_in[0];
  const float* W5a  = (const float*)d_in[1];
  const float* g5a  = (const float*)d_in[2];
  const float* b5a  = (const float*)d_in[3];
  const float* W5c  = (const float*)d_in[4];
  const float* g5c  = (const float*)d_in[5];
  const float* b5c  = (const float*)d_in[6];
  const float* Wq   = (const float*)d_in[7];
  const float* bq   = (const float*)d_in[8];
  const float* Wk   = (const float*)d_in[9];
  const float* bk   = (const float*)d_in[10];
  const float* Wv   = (const float*)d_in[11];
  const float* bv   = (const float*)d_in[12];
  const float* gpam = (const float*)d_in[13];
  const float* gcam = (const float*)d_in[14];
  const float* W51  = (const float*)d_in[15];
  const float* g51  = (const float*)d_in[16];
  const float* b51  = (const float*)d_in[17];
  const float* W52  = (const float*)d_in[18];
  const float* g52  = (const float*)d_in[19];
  const float* b52  = (const float*)d_in[20];
  const float* W6   = (const float*)d_in[21];
  const float* b6   = (const float*)d_in[22];
  const float* W7   = (const float*)d_in[23];
  const float* b7   = (const float*)d_in[24];
  const float* W8   = (const float*)d_in[25];
  const float* b8   = (const float*)d_in[26];

  float* ws = (float*)d_ws;
  const size_t T = (size_t)BB * IC * LP;          // 1,048,576 floats
  float* ypad1 = ws;
  float* ypad2 = ypad1 + T;
  float* feat1 = ypad2 + T;
  float* feat2 = feat1 + T;
  float* qkbuf = feat2 + T;                       // BB*16*LP
  float* vbuf  = qkbuf + (size_t)BB * 16 * LP;
  float* saf   = vbuf + T;
  float* scf   = saf + T;
  float* za    = scf + T;                         // conv3 out -> sa_conv (in place)
  float* zb    = za + T;                          // conv3 out -> sc_conv (in place)
  float* coef  = zb + T;                          // 4 layers * 64 * 2
  float* Wqk   = coef + 512;                      // 16x64
  float* bqk   = Wqk + 1024;                      // 16
  float* camE  = bqk + 16;                        // BB*64*64
  float* Sbuf  = camE + (size_t)BB * IC * IC;     // BB*IC

  float* out_sasc = (float*)d_out;                          // [8,128]
  float* out_sa   = out_sasc + BB * OCH;                    // [8,128,2048]
  float* out_sc   = out_sa + (size_t)BB * OCH * LP;         // [8,128,2048]

  dim3 blk(128);
  init_pack_kernel<<<16, 256, 0, stream>>>(Wq, Wk, bq, bk, ypad1, ypad2, Wqk, bqk);

  // branch 1+2 front convs (into padded buffers), BN stats, apply
  conv1x1_wmma_kernel<<<dim3(32, 4, BB), blk, 0, stream>>>(x, W5a, nullptr, ypad1, CIN, LIN, LIN, LP, 1);
  conv1x1_wmma_kernel<<<dim3(32, 4, BB), blk, 0, stream>>>(x, W5c, nullptr, ypad2, CIN, LIN, LIN, LP, 1);
  bn_stats_kernel<<<IC, 256, 0, stream>>>(ypad1, g5a, b5a, coef);
  bn_stats_kernel<<<IC, 256, 0, stream>>>(ypad2, g5c, b5c, coef + 128);
  bn_apply_kernel<<<1024, 256, 0, stream>>>(ypad1, coef, feat1);
  bn_apply_kernel<<<1024, 256, 0, stream>>>(ypad2, coef + 128, feat2);

  // PAM: q|k packed conv, v conv, fused flash attention
  conv1x1_wmma_kernel<<<dim3(32, 1, BB), blk, 0, stream>>>(feat1, Wqk, bqk, qkbuf, IC, LP, LP, LP, 0);
  conv1x1_wmma_kernel<<<dim3(32, 4, BB), blk, 0, stream>>>(feat1, Wv, bv, vbuf, IC, LP, LP, LP, 0);
  pam_flash_kernel<<<dim3(LP / 16, BB), blk, 0, stream>>>(qkbuf, vbuf, feat1, gpam, saf);

  // CAM
  cam_energy_kernel<<<dim3(4, BB), 256, 0, stream>>>(feat2, camE);
  cam_softmax_kernel<<<BB, IC, 0, stream>>>(camE);
  cam_out_kernel<<<4096, 256, 0, stream>>>(camE, feat2, gcam, scf);

  // conv3 + BN + relu for both branches
  conv3_kernel<<<dim3(LP / 256, IC, BB), 256, 0, stream>>>(saf, W51, za);
  conv3_kernel<<<dim3(LP / 256, IC, BB), 256, 0, stream>>>(scf, W52, zb);
  bn_stats_kernel<<<IC, 256, 0, stream>>>(za, g51, b51, coef + 256);
  bn_stats_kernel<<<IC, 256, 0, stream>>>(zb, g52, b52, coef + 384);
  bn_apply_kernel<<<1024, 256, 0, stream>>>(za, coef + 256, za);
  bn_apply_kernel<<<1024, 256, 0, stream>>>(zb, coef + 384, zb);

  // heads
  conv1x1_wmma_kernel<<<dim3(32, 8, BB), blk, 0, stream>>>(za, W6, b6, out_sa, IC, LP, LP, LP, 0);
  conv1x1_wmma_kernel<<<dim3(32, 8, BB), blk, 0, stream>>>(zb, W7, b7, out_sc, IC, LP, LP, LP, 0);
  sum_l_kernel<<<BB * IC, 256, 0, stream>>>(za, zb, Sbuf);
  final_kernel<<<BB, OCH, 0, stream>>>(W8, b8, Sbuf, out_sasc);
}